// EdgeBasedVertexAveraging_52106543235552
// MI455X (gfx1250) — compile-verified
//
#include <hip/hip_runtime.h>

typedef __attribute__((ext_vector_type(2))) float v2f;
typedef __attribute__((ext_vector_type(8))) float v8f;
typedef __attribute__((ext_vector_type(4))) unsigned int v4u;

#define ST_UNK  0
#define ST_HEAD 1
#define ST_NONH 2

// ---------------------------------------------------------------------------
// Phase 0: zero the means region of d_out, init MIS status.
// ---------------------------------------------------------------------------
__global__ void init_kernel(int* __restrict__ status, float* __restrict__ out_means,
                            int N, long long total) {
  long long idx = (long long)blockIdx.x * blockDim.x + threadIdx.x;
  if (idx < total) out_means[idx] = 0.0f;
  if (idx < N) status[idx] = ST_UNK;
}

// ---------------------------------------------------------------------------
// Phase 1: one fixed-point step of lexicographically-first MIS.
// Monotone in-place updates (UNK -> HEAD/NONH only) => confluent/deterministic.
// Rounds needed = longest increasing adjacent path (<50 w.h.p. for this graph).
// ---------------------------------------------------------------------------
__global__ void mis_step(const int* __restrict__ nbr, const int* __restrict__ deg,
                         int* __restrict__ status, int N, int maxdeg) {
  int i = blockIdx.x * blockDim.x + threadIdx.x;
  if (i >= N) return;
  if (status[i] != ST_UNK) return;
  const int* row = nbr + (size_t)i * maxdeg;
  int d = deg[i];
  bool any_unk = false;
  for (int s = 0; s < d; ++s) {
    int j = row[s];
    if (j < i) {
      int st = status[j];
      if (st == ST_HEAD) { status[i] = ST_NONH; return; }
      if (st == ST_UNK) any_unk = true;
    }
  }
  if (!any_unk) status[i] = ST_HEAD;
}

// ---------------------------------------------------------------------------
// Phase 2: single-block scan over head flags.
// ---------------------------------------------------------------------------
__global__ void scan_heads(const int* __restrict__ status, int* __restrict__ headIdx,
                           int* __restrict__ numC, float* __restrict__ out_mask,
                           float* __restrict__ out_nc, int N) {
  __shared__ int part[1024];
  int t = threadIdx.x;
  int per = (N + (int)blockDim.x - 1) / (int)blockDim.x;
  int base = t * per;
  int c = 0;
  for (int k = 0; k < per; ++k) {
    int i = base + k;
    if (i < N && status[i] == ST_HEAD) c++;
  }
  part[t] = c;
  __syncthreads();
  for (int off = 1; off < (int)blockDim.x; off <<= 1) {
    int v = (t >= off) ? part[t - off] : 0;
    __syncthreads();
    part[t] += v;
    __syncthreads();
  }
  int run = part[t] - c;
  for (int k = 0; k < per; ++k) {
    int i = base + k;
    if (i < N) {
      bool h = (status[i] == ST_HEAD);
      out_mask[i] = h ? 1.0f : 0.0f;
      if (h) { headIdx[run] = i; run++; }
    }
  }
  if (t == (int)blockDim.x - 1) {
    numC[0] = part[t];
    out_nc[0] = (float)part[t];
  }
}

// ---------------------------------------------------------------------------
// Phase 3: each non-head is owned by its minimum-index head neighbor.
// ---------------------------------------------------------------------------
__global__ void assign_owner(const int* __restrict__ nbr, const int* __restrict__ deg,
                             const int* __restrict__ status, int* __restrict__ own,
                             int N, int maxdeg) {
  int k = blockIdx.x * blockDim.x + threadIdx.x;
  if (k >= N) return;
  if (status[k] == ST_HEAD) { own[k] = k; return; }
  const int* row = nbr + (size_t)k * maxdeg;
  int d = deg[k];
  int m = 0x7fffffff;
  for (int s = 0; s < d; ++s) {
    int j = row[s];
    if (status[j] == ST_HEAD && j < m) m = j;
  }
  own[k] = m;
}

// ---------------------------------------------------------------------------
// WMMA slot permutation: within each 8-K chunk, lanes 0-15 (K-halves 0,1 of
// the two WMMAs) read slots 0..3 = K{0,1,4,5}; lanes 16-31 read slots 4..7 =
// K{2,3,6,7}. One ds_load_b128 per lane per chunk.
// ---------------------------------------------------------------------------
__device__ __forceinline__ int slot_of(int k) {
  int r = k & 7;
  int t = r >> 1;
  if (t == 1 || t == 2) r ^= 6;  // 2<->4, 3<->5
  return (k & ~7) | r;
}

// ---------------------------------------------------------------------------
// Phase 4: cluster means via V_WMMA_F32_16X16X4_F32 (full f32 precision).
// Block = 16 clusters; 4 waves, wave w covers column tiles {w, w+4, w+8, w+12}.
// packed[k] = vtx*D*4 | ownerCode (D*4 = 1024 => low 10 bits free; pad=0x3FF).
// ---------------------------------------------------------------------------
__global__ void __launch_bounds__(128)
cluster_means(const float* __restrict__ vertices, const int* __restrict__ nbr,
              const int* __restrict__ deg, const int* __restrict__ headIdx,
              const int* __restrict__ own, const int* __restrict__ numC,
              float* __restrict__ out, int N, int D, int maxdeg, int KMAX) {
  extern __shared__ __align__(16) unsigned int packed[];  // [KMAX]
  __shared__ int   cnt[16];
  __shared__ int   off[16];
  __shared__ float recip[16];
  __shared__ int   s_R, s_ncl;

  const int tid = threadIdx.x;
  const int c0 = blockIdx.x * 16;

  if (tid == 0) {
    int nc = numC[0];
    int ncl = nc - c0;
    ncl = ncl < 0 ? 0 : (ncl > 16 ? 16 : ncl);
    s_ncl = ncl;
  }
  __syncthreads();
  const int ncl = s_ncl;
  if (ncl == 0) return;  // uniform across block

  // count members per cluster
  if (tid < 16) {
    int c = 0;
    if (tid < ncl) {
      int i = headIdx[c0 + tid];
      const int* row = nbr + (size_t)i * maxdeg;
      int d = deg[i];
      c = 1;
      for (int s = 0; s < d; ++s)
        if (own[row[s]] == i) c++;
    }
    cnt[tid] = c;
    recip[tid] = c ? 1.0f / (float)c : 0.0f;
  }
  __syncthreads();
  if (tid == 0) {
    int r = 0;
    for (int m = 0; m < 16; ++m) { off[m] = r; r += cnt[m]; }
    int Rp = (r + 7) & ~7;  // pad K to multiple of 8 (two WMMAs per chunk)
    for (int p = r; p < Rp; ++p) packed[slot_of(p)] = 0x3FFu;
    s_R = Rp;
  }
  __syncthreads();
  const unsigned rowBytes = (unsigned)D * 4u;  // 1024 for D=256
  if (tid < ncl) {
    int i = headIdx[c0 + tid];
    const int* row = nbr + (size_t)i * maxdeg;
    int d = deg[i];
    int o = off[tid];
    packed[slot_of(o)] = (unsigned)i * rowBytes | (unsigned)tid;
    ++o;
    for (int s = 0; s < d; ++s) {
      int k = row[s];
      if (own[k] == i) {
        packed[slot_of(o)] = (unsigned)k * rowBytes | (unsigned)tid;
        ++o;
      }
    }
  }
  __syncthreads();

  const int R    = s_R;
  const int lane = tid & 31;
  const int wv   = tid >> 5;
  const int lm   = lane & 15;
  const unsigned ow = (unsigned)lm;
  const int hi   = lane >> 4;     // 0: K{0,1,4,5}; 1: K{2,3,6,7}
  const int half8 = hi * 8;       // D rows v / v+8 split
  const char* vb = (const char*)vertices;

  for (int ct = wv; ct < 16; ct += 4) {
    const unsigned colb = (unsigned)(ct * 16 + lm) * 4u;
    v8f acc = {0.f, 0.f, 0.f, 0.f, 0.f, 0.f, 0.f, 0.f};
#pragma unroll 2
    for (int kc = 0; kc < R; kc += 8) {
      v4u q = *(const v4u*)&packed[kc + hi * 4];  // ds_load_b128
      float b0 = *(const float*)(vb + (size_t)((q.x & ~1023u) + colb));
      float b1 = *(const float*)(vb + (size_t)((q.y & ~1023u) + colb));
      float b2 = *(const float*)(vb + (size_t)((q.z & ~1023u) + colb));
      float b3 = *(const float*)(vb + (size_t)((q.w & ~1023u) + colb));
      v2f A0, B0, A1, B1;
      A0.x = ((q.x & 1023u) == ow) ? 1.0f : 0.0f;
      A0.y = ((q.y & 1023u) == ow) ? 1.0f : 0.0f;
      A1.x = ((q.z & 1023u) == ow) ? 1.0f : 0.0f;
      A1.y = ((q.w & 1023u) == ow) ? 1.0f : 0.0f;
      B0.x = b0; B0.y = b1;
      B1.x = b2; B1.y = b3;
      acc = __builtin_amdgcn_wmma_f32_16x16x4_f32(
          false, A0, false, B0, (short)0, acc, false, false);
      acc = __builtin_amdgcn_wmma_f32_16x16x4_f32(
          false, A1, false, B1, (short)0, acc, false, false);
    }
#pragma unroll
    for (int v = 0; v < 8; ++v) {
      int m = v + half8;
      if (m < ncl)
        out[(size_t)(c0 + m) * D + ct * 16 + lm] = acc[v] * recip[m];
    }
  }
}

// ---------------------------------------------------------------------------
extern "C" void kernel_launch(void* const* d_in, const int* in_sizes, int n_in,
                              void* d_out, int out_size, void* d_ws, size_t ws_size,
                              hipStream_t stream) {
  const float* vertices = (const float*)d_in[0];
  const int*   nbr      = (const int*)d_in[1];
  const int*   deg      = (const int*)d_in[2];

  int N      = in_sizes[2];
  int D      = in_sizes[0] / N;
  int maxdeg = in_sizes[1] / N;

  float* out_means = (float*)d_out;
  float* out_mask  = out_means + (size_t)N * D;
  float* out_nc    = out_mask + N;

  int* status  = (int*)d_ws;
  int* headIdx = status + N;
  int* own     = headIdx + N;
  int* numC    = own + N;

  long long total = (long long)N * D;
  dim3 blk(256);
  dim3 grid_init((unsigned)((total + 255) / 256));
  init_kernel<<<grid_init, blk, 0, stream>>>(status, out_means, N, total);

  dim3 grid_n((N + 255) / 256);
  for (int it = 0; it < 64; ++it)
    mis_step<<<grid_n, blk, 0, stream>>>(nbr, deg, status, N, maxdeg);

  scan_heads<<<1, 1024, 0, stream>>>(status, headIdx, numC, out_mask, out_nc, N);

  assign_owner<<<grid_n, blk, 0, stream>>>(nbr, deg, status, own, N, maxdeg);

  int KMAX = ((16 * (maxdeg + 1)) + 7) & ~7;
  size_t shmem = (size_t)KMAX * sizeof(unsigned int);
  dim3 grid_means((N + 15) / 16);
  cluster_means<<<grid_means, dim3(128), shmem, stream>>>(
      vertices, nbr, deg, headIdx, own, numC, out_means, N, D, maxdeg, KMAX);
}